// DOTGATConvDGL_39032662786145
// MI455X (gfx1250) — compile-verified
//
#include <hip/hip_runtime.h>
#include <math.h>

typedef __attribute__((ext_vector_type(2))) float v2f;
typedef __attribute__((ext_vector_type(8))) float v8f;

#define IN_SIZE 128
#define NUM_HEADS 4
#define OUT_SIZE 32
#define HID (NUM_HEADS * OUT_SIZE)   // 128

// ---------------------------------------------------------------------------
// Kernel 1: h = feat @ W  via V_WMMA_F32_16X16X4_F32 (f32, exact numerics).
// One wave per 16x16 output tile; K=128 in steps of 4 -> 32 WMMA per tile.
// A frag (16x4 f32, 2 VGPRs): lane<16 -> (M=lane, K=k0,k0+1); lane>=16 ->
// (M=lane-16, K=k0+2,k0+3).  B frag (4x16) symmetric.  C/D: VGPR r holds
// (M = r + 8*(lane>=16), N = lane%16).
// ---------------------------------------------------------------------------
__global__ void proj_wmma(const float* __restrict__ feat,
                          const float* __restrict__ W,
                          float* __restrict__ h, int nNodes) {
  const int waveInBlock = threadIdx.x >> 5;
  const int lane        = threadIdx.x & 31;
  const int tile        = blockIdx.x * 8 + waveInBlock;
  const int mTiles      = (nNodes + 15) >> 4;
  const int mt = tile >> 3;   // 128/16 = 8 N-tiles
  const int nt = tile & 7;
  if (mt >= mTiles) return;
  const int m0   = mt << 4;
  const int n0   = nt << 4;
  const int half = lane >> 4;   // 0: K=k0..k0+1, 1: K=k0+2..k0+3
  const int l16  = lane & 15;

  v8f c = {};
  for (int k0 = 0; k0 < IN_SIZE; k0 += 4) {
    const int ka = k0 + (half << 1);
    int arow = m0 + l16;
    if (arow >= nNodes) arow = nNodes - 1;   // clamp (tail tiles)
    v2f a, b;
    a.x = feat[(size_t)arow * IN_SIZE + ka];
    a.y = feat[(size_t)arow * IN_SIZE + ka + 1];
    b.x = W[(size_t)ka       * HID + n0 + l16];
    b.y = W[(size_t)(ka + 1) * HID + n0 + l16];
    c = __builtin_amdgcn_wmma_f32_16x16x4_f32(
        /*neg_a=*/false, a, /*neg_b=*/false, b,
        /*c_mod=*/(short)0, c, /*reuse_a=*/false, /*reuse_b=*/false);
  }
  for (int r = 0; r < 8; ++r) {
    const int row = m0 + r + (half << 3);
    if (row < nNodes) h[(size_t)row * HID + n0 + l16] = c[r];
  }
}

// ---------------------------------------------------------------------------
// Kernel 2: CSR row pointers from the sorted dst array via binary search.
// rp[n] = first edge index with dst >= n;  rp[N] = E.
// ---------------------------------------------------------------------------
__global__ void build_rowptr(const int* __restrict__ dst, int* __restrict__ rp,
                             int nNodes, int E) {
  const int n = blockIdx.x * blockDim.x + threadIdx.x;
  if (n > nNodes) return;
  int lo = 0, hi = E;
  while (lo < hi) {
    const int mid = (lo + hi) >> 1;
    if (dst[mid] < n) lo = mid + 1; else hi = mid;
  }
  rp[n] = lo;
}

// ---------------------------------------------------------------------------
// Kernel 3: per-edge attention scores.  One wave per edge.  Lane l loads the
// float4 slice [4l,4l+4) of h_src/h_dst (whole 512B rows, L2-resident),
// reduces the per-head partial dot across its 8-lane head group.
// ---------------------------------------------------------------------------
__global__ void edge_scores(const float* __restrict__ h,
                            const int* __restrict__ src,
                            const int* __restrict__ dst,
                            float* __restrict__ esc, int E) {
  const int wid  = (int)((blockIdx.x * (size_t)blockDim.x + threadIdx.x) >> 5);
  const int lane = threadIdx.x & 31;
  if (wid >= E) return;
  const int s = src[wid], d = dst[wid];
  const float4 a = ((const float4*)(h + (size_t)s * HID))[lane];
  const float4 b = ((const float4*)(h + (size_t)d * HID))[lane];
  float p = a.x * b.x + a.y * b.y + a.z * b.z + a.w * b.w;
  p += __shfl_xor(p, 1, 32);
  p += __shfl_xor(p, 2, 32);
  p += __shfl_xor(p, 4, 32);          // lanes 8h..8h+7 hold head-h dot
  if ((lane & 7) == 0)
    esc[(size_t)wid * NUM_HEADS + (lane >> 3)] = p * 0.17677669529663689f; // /sqrt(32)
}

// ---------------------------------------------------------------------------
// Kernel 4: one-pass online softmax + weighted aggregation.  One wave per
// destination node (edges are contiguous thanks to sorted dst).  Lane l owns
// out slice [4l,4l+4) (head = l/8); per-head running max m, running sum s,
// rescaled accumulator — single sweep over the node's edges, no atomics.
// ---------------------------------------------------------------------------
__global__ void node_aggregate(const float* __restrict__ h,
                               const int* __restrict__ src,
                               const float* __restrict__ esc,
                               const int* __restrict__ rp,
                               float* __restrict__ out, int nNodes) {
  const int wid  = (int)((blockIdx.x * (size_t)blockDim.x + threadIdx.x) >> 5);
  const int lane = threadIdx.x & 31;
  if (wid >= nNodes) return;
  const int beg = rp[wid], end = rp[wid + 1];
  const int head = lane >> 3;

  float m = -INFINITY, ssum = 0.f;
  float4 acc = {0.f, 0.f, 0.f, 0.f};
  for (int i = beg; i < end; ++i) {
    const int s = src[i];
    const float e = esc[(size_t)i * NUM_HEADS + head];
    const float4 hs = ((const float4*)(h + (size_t)s * HID))[lane];
    const float mn    = fmaxf(m, e);
    const float scale = __expf(m - mn);   // first iter: exp(-inf)=0 -> clean init
    const float w     = __expf(e - mn);
    ssum  = ssum * scale + w;
    acc.x = acc.x * scale + w * hs.x;
    acc.y = acc.y * scale + w * hs.y;
    acc.z = acc.z * scale + w * hs.z;
    acc.w = acc.w * scale + w * hs.w;
    m = mn;
  }
  const float inv = (end > beg) ? (1.0f / ssum) : 0.0f;
  float4 o = {acc.x * inv, acc.y * inv, acc.z * inv, acc.w * inv};
  ((float4*)(out + (size_t)wid * HID))[lane] = o;
}

// ---------------------------------------------------------------------------
extern "C" void kernel_launch(void* const* d_in, const int* in_sizes, int n_in,
                              void* d_out, int out_size, void* d_ws, size_t ws_size,
                              hipStream_t stream) {
  const float* feat = (const float*)d_in[0];
  const int*   src  = (const int*)d_in[1];
  const int*   dst  = (const int*)d_in[2];
  const float* W    = (const float*)d_in[3];
  const int nNodes  = in_sizes[0] / IN_SIZE;
  const int E       = in_sizes[1];
  float* out = (float*)d_out;

  // Workspace layout: h [N*128 f32] | esc [E*4 f32] | rp [N+1 i32]
  char*  ws  = (char*)d_ws;
  float* h   = (float*)ws;
  float* esc = (float*)(ws + (size_t)nNodes * HID * sizeof(float));
  int*   rp  = (int*)  (ws + (size_t)nNodes * HID * sizeof(float)
                           + (size_t)E * NUM_HEADS * sizeof(float));

  const int mTiles = (nNodes + 15) / 16;
  const int tiles  = mTiles * (HID / 16);          // 8 N-tiles
  proj_wmma     <<<(tiles + 7) / 8,        256, 0, stream>>>(feat, W, h, nNodes);
  build_rowptr  <<<(nNodes + 1 + 255)/256, 256, 0, stream>>>(dst, rp, nNodes, E);
  edge_scores   <<<(E + 7) / 8,            256, 0, stream>>>(h, src, dst, esc, E);
  node_aggregate<<<(nNodes + 7) / 8,       256, 0, stream>>>(h, src, esc, rp, out, nNodes);
}